// SoftRotationSolver_53188874994219
// MI455X (gfx1250) — compile-verified
//
#include <hip/hip_runtime.h>
#include <hip/hip_bf16.h>

#define BB   64
#define NN   2048
#define EPSV 1e-8f

typedef float v2f __attribute__((ext_vector_type(2)));
typedef float v8f __attribute__((ext_vector_type(8)));

// ---------------------------------------------------------------------------
// Kernel 0: zero the per-batch H accumulator (64 x 9 floats) in workspace.
// ---------------------------------------------------------------------------
__global__ void __launch_bounds__(256) srs_zeroH_kernel(float* Hbuf) {
    int i = blockIdx.x * blockDim.x + threadIdx.x;
    if (i < BB * 9) Hbuf[i] = 0.0f;
}

// ---------------------------------------------------------------------------
// Kernel 1: streaming pass over P.
//   Per wave: 16-row band. WMMA f32 16x16x4 computes Q = P_tile * [db | 1 | 0..]
//   accumulated over K=2048. Epilogue: W = Q/(s+eps), H[i][j] += a_i * W_j,
//   atomically accumulated into Hbuf[batch][3][3].
// Grid: B * (N/64) blocks of 128 threads (4 waves, 64 rows per block).
// ---------------------------------------------------------------------------
__global__ void __launch_bounds__(128) srs_accumH_kernel(
    const float* __restrict__ P,
    const float* __restrict__ dirs_a,
    const float* __restrict__ dirs_b,
    float* __restrict__ Hbuf) {

    __shared__ float s_db[NN * 4 + 4];   // [m][0..2]=dirs_b, [m][3]=1.0, +4 zero pad

    const int tid   = threadIdx.x;
    const int batch = blockIdx.x >> 5;   // 32 row-blocks per batch
    const int rblk  = blockIdx.x & 31;   // 64 rows per block

    // Stage extended dirs_b into LDS (plus zero pad words).
    for (int idx = tid; idx < NN * 4 + 4; idx += 128) {
        float v = 0.0f;
        if (idx < NN * 4) {
            int m = idx >> 2, jj = idx & 3;
            v = (jj < 3) ? dirs_b[((size_t)batch * NN + m) * 3 + jj] : 1.0f;
        }
        s_db[idx] = v;
    }
    __syncthreads();

    const int wave    = tid >> 5;
    const int lane    = tid & 31;
    const int j       = lane & 15;        // B/C column index
    const int kOff    = (lane >> 4) << 1; // A/B K offset: 0 or 2
    const int rowBase = rblk * 64 + wave * 16;
    const int row     = rowBase + j;      // A-matrix row M = lane&15
    const float* pRow = P + ((size_t)batch * NN + row) * NN;

    // Branchless B-fragment addressing: columns >=4 read the zero pad (stride 0).
    int bBase, bStride;
    if (j < 4) { bBase = j;               bStride = 4; }
    else       { bBase = NN * 4 + (j & 3); bStride = 0; }

    v8f c0 = {};
    v8f c1 = {};

    for (int cb = 0; cb < NN; cb += 16) {
        const int m0 = cb + kOff;
        v2f a0 = *(const v2f*)(pRow + m0);
        v2f a1 = *(const v2f*)(pRow + m0 + 4);
        v2f a2 = *(const v2f*)(pRow + m0 + 8);
        v2f a3 = *(const v2f*)(pRow + m0 + 12);
        __builtin_prefetch(pRow + m0 + 128, 0, 1);   // global_prefetch_b8, stream ahead

        v2f b0 = { s_db[bBase + (m0     ) * bStride], s_db[bBase + (m0 +  1) * bStride] };
        v2f b1 = { s_db[bBase + (m0 +  4) * bStride], s_db[bBase + (m0 +  5) * bStride] };
        v2f b2 = { s_db[bBase + (m0 +  8) * bStride], s_db[bBase + (m0 +  9) * bStride] };
        v2f b3 = { s_db[bBase + (m0 + 12) * bStride], s_db[bBase + (m0 + 13) * bStride] };

        c0 = __builtin_amdgcn_wmma_f32_16x16x4_f32(false, a0, false, b0, (short)0, c0, false, false);
        c1 = __builtin_amdgcn_wmma_f32_16x16x4_f32(false, a1, false, b1, (short)0, c1, false, false);
        c0 = __builtin_amdgcn_wmma_f32_16x16x4_f32(false, a2, false, b2, (short)0, c0, false, false);
        c1 = __builtin_amdgcn_wmma_f32_16x16x4_f32(false, a3, false, b3, (short)0, c1, false, false);
    }

    union { v8f v; float f[8]; } cc;
    cc.v = c0 + c1;

    // C layout: lane holds column j, rows (v + 8*(lane>=16)) of the band.
    // Row sum s lives in column 3 -> lane (half|3).
    const int srcS = (lane & 16) | 3;
    float hp0 = 0.0f, hp1 = 0.0f, hp2 = 0.0f;
    #pragma unroll
    for (int v = 0; v < 8; ++v) {
        float q = cc.f[v];
        float s = __shfl(q, srcS, 32);     // all lanes execute the shuffle
        float w = q / (s + EPSV);
        int n = rowBase + v + ((lane & 16) ? 8 : 0);
        if (j < 3) {
            const float* ap = dirs_a + ((size_t)batch * NN + n) * 3;
            hp0 += ap[0] * w;
            hp1 += ap[1] * w;
            hp2 += ap[2] * w;
        }
    }
    // Combine the two row halves (lane L with lane L^16, same column j).
    hp0 += __shfl_xor(hp0, 16, 32);
    hp1 += __shfl_xor(hp1, 16, 32);
    hp2 += __shfl_xor(hp2, 16, 32);

    if (lane < 3) {
        float* hb = Hbuf + batch * 9;
        atomicAdd(hb + 0 * 3 + lane, hp0);
        atomicAdd(hb + 1 * 3 + lane, hp1);
        atomicAdd(hb + 2 * 3 + lane, hp2);
    }
}

// ---------------------------------------------------------------------------
// Kernel 2: per-batch 3x3 SVD via cyclic Jacobi on H^T H, then R = Vh U^T with
// the reference's reflection fix (flip last COLUMN of Vh == last row of V).
// ---------------------------------------------------------------------------
__device__ __forceinline__ float srs_det3(const float M[3][3]) {
    return M[0][0] * (M[1][1] * M[2][2] - M[1][2] * M[2][1])
         - M[0][1] * (M[1][0] * M[2][2] - M[1][2] * M[2][0])
         + M[0][2] * (M[1][0] * M[2][1] - M[1][1] * M[2][0]);
}

__global__ void __launch_bounds__(64) srs_svd_kernel(const float* __restrict__ Hbuf,
                                                     float* __restrict__ Rout) {
    const int b = threadIdx.x;
    if (b >= BB) return;

    float H[3][3];
    #pragma unroll
    for (int i = 0; i < 3; ++i)
        #pragma unroll
        for (int jj = 0; jj < 3; ++jj)
            H[i][jj] = Hbuf[b * 9 + i * 3 + jj];

    // A = H^T H (symmetric PSD)
    float A[3][3];
    #pragma unroll
    for (int i = 0; i < 3; ++i)
        #pragma unroll
        for (int jj = 0; jj < 3; ++jj)
            A[i][jj] = H[0][i] * H[0][jj] + H[1][i] * H[1][jj] + H[2][i] * H[2][jj];

    float V[3][3] = {{1.f,0.f,0.f},{0.f,1.f,0.f},{0.f,0.f,1.f}};

    const int PPr[3] = {0, 0, 1};
    const int QQr[3] = {1, 2, 2};
    for (int sw = 0; sw < 10; ++sw) {
        #pragma unroll
        for (int r = 0; r < 3; ++r) {
            const int p = PPr[r], q = QQr[r];
            float apq = A[p][q];
            if (fabsf(apq) > 1e-30f) {
                float theta = (A[q][q] - A[p][p]) / (2.0f * apq);
                float t   = copysignf(1.0f, theta) / (fabsf(theta) + sqrtf(theta * theta + 1.0f));
                float cth = 1.0f / sqrtf(t * t + 1.0f);
                float sth = t * cth;
                #pragma unroll
                for (int k = 0; k < 3; ++k) {          // A <- A * G
                    float akp = A[k][p], akq = A[k][q];
                    A[k][p] = cth * akp - sth * akq;
                    A[k][q] = sth * akp + cth * akq;
                }
                #pragma unroll
                for (int k = 0; k < 3; ++k) {          // A <- G^T * A
                    float apk = A[p][k], aqk = A[q][k];
                    A[p][k] = cth * apk - sth * aqk;
                    A[q][k] = sth * apk + cth * aqk;
                }
                #pragma unroll
                for (int k = 0; k < 3; ++k) {          // V <- V * G
                    float vkp = V[k][p], vkq = V[k][q];
                    V[k][p] = cth * vkp - sth * vkq;
                    V[k][q] = sth * vkp + cth * vkq;
                }
            }
        }
    }

    float lam[3] = {A[0][0], A[1][1], A[2][2]};
    // sort descending (swap V columns along)
    #pragma unroll
    for (int pass = 0; pass < 3; ++pass) {
        const int i1 = (pass == 1) ? 1 : 0;
        const int i2 = (pass == 0) ? 1 : 2;
        if (lam[i1] < lam[i2]) {
            float tl = lam[i1]; lam[i1] = lam[i2]; lam[i2] = tl;
            #pragma unroll
            for (int k = 0; k < 3; ++k) {
                float tv = V[k][i1]; V[k][i1] = V[k][i2]; V[k][i2] = tv;
            }
        }
    }

    // U columns: u_k = H v_k / s_k
    float U[3][3];
    #pragma unroll
    for (int k = 0; k < 3; ++k) {
        float s  = sqrtf(fmaxf(lam[k], 0.0f));
        float t0 = H[0][0] * V[0][k] + H[0][1] * V[1][k] + H[0][2] * V[2][k];
        float t1 = H[1][0] * V[0][k] + H[1][1] * V[1][k] + H[1][2] * V[2][k];
        float t2 = H[2][0] * V[0][k] + H[2][1] * V[1][k] + H[2][2] * V[2][k];
        if (s > 1e-12f) {
            float inv = 1.0f / s;
            U[0][k] = t0 * inv; U[1][k] = t1 * inv; U[2][k] = t2 * inv;
        } else if (k == 2) {
            float cx = U[1][0] * U[2][1] - U[2][0] * U[1][1];
            float cy = U[2][0] * U[0][1] - U[0][0] * U[2][1];
            float cz = U[0][0] * U[1][1] - U[1][0] * U[0][1];
            float rn = 1.0f / sqrtf(cx * cx + cy * cy + cz * cz + 1e-30f);
            U[0][2] = cx * rn; U[1][2] = cy * rn; U[2][2] = cz * rn;
        } else {
            U[0][k] = (k == 0) ? 1.0f : 0.0f;
            U[1][k] = (k == 1) ? 1.0f : 0.0f;
            U[2][k] = 0.0f;
        }
    }

    // det(R) = det(Vh) * det(U^T) = det(V) * det(U)
    float d = srs_det3(U) * srs_det3(V);
    if (d < 0.0f) {       // flip last column of Vh == last row of V
        V[2][0] = -V[2][0]; V[2][1] = -V[2][1]; V[2][2] = -V[2][2];
    }

    // R = Vh * U^T : R[i][j] = sum_k V[k][i] * U[j][k]
    #pragma unroll
    for (int i = 0; i < 3; ++i)
        #pragma unroll
        for (int jj = 0; jj < 3; ++jj) {
            float rv = V[0][i] * U[jj][0] + V[1][i] * U[jj][1] + V[2][i] * U[jj][2];
            Rout[b * 9 + i * 3 + jj] = rv;
        }
}

// ---------------------------------------------------------------------------
extern "C" void kernel_launch(void* const* d_in, const int* in_sizes, int n_in,
                              void* d_out, int out_size, void* d_ws, size_t ws_size,
                              hipStream_t stream) {
    const float* P      = (const float*)d_in[0];
    const float* dirs_a = (const float*)d_in[1];
    const float* dirs_b = (const float*)d_in[2];
    float* Hbuf = (float*)d_ws;              // 64*9 floats
    float* Rout = (float*)d_out;             // 64*3*3 floats

    srs_zeroH_kernel<<<dim3(3), 256, 0, stream>>>(Hbuf);
    srs_accumH_kernel<<<dim3(BB * (NN / 64)), 128, 0, stream>>>(P, dirs_a, dirs_b, Hbuf);
    srs_svd_kernel<<<dim3(1), 64, 0, stream>>>(Hbuf, Rout);
}